// GNN_34041910788167
// MI455X (gfx1250) — compile-verified
//
#include <hip/hip_runtime.h>
#include <hip/hip_bf16.h>

typedef float v2f __attribute__((ext_vector_type(2)));
typedef float v8f __attribute__((ext_vector_type(8)));

#define FD 128   // feature dim (in & out of both layers)

// ---------------- degree / normalization ----------------

__global__ void gcn_deg_kernel(const int* __restrict__ dst, float* __restrict__ deg, int E) {
    int e = blockIdx.x * blockDim.x + threadIdx.x;
    if (e < E) unsafeAtomicAdd(&deg[dst[e]], 1.0f);
}

__global__ void gcn_dis_kernel(const float* __restrict__ deg, float* __restrict__ dis, int N) {
    int i = blockIdx.x * blockDim.x + threadIdx.x;
    if (i < N) dis[i] = rsqrtf(deg[i] + 1.0f);   // +1 = self loop; deg>=1 always
}

// ---------------- dense GEMM via V_WMMA_F32_16X16X4_F32 ----------------
// C[M x 128] = A[M x 128] * W[128 x 128]; one wave computes one 16x16 tile.
// A-frag (16x4 f32, 2 VGPRs): lanes 0-15 -> M=lane, K = k+0 / k+1
//                             lanes 16-31 -> M=lane-16, K = k+2 / k+3
// B-frag (4x16 f32, 2 VGPRs): lanes 0-15 -> row k+0 / k+1 ; lanes 16-31 -> row k+2 / k+3
// C/D (8 VGPRs): VGPR v -> M = v (lanes 0-15) or v+8 (lanes 16-31), N = lane%16

__global__ void gcn_gemm_wmma(const float* __restrict__ A, const float* __restrict__ W,
                              float* __restrict__ C, int M) {
    const int K  = FD;
    const int Nc = FD;
    int wave = (int)((blockIdx.x * blockDim.x + threadIdx.x) >> 5);
    int lane = threadIdx.x & 31;
    const int tiles_n = Nc / 16;                 // 8
    int tile_m = wave / tiles_n;
    int tile_n = wave - tile_m * tiles_n;
    if (tile_m * 16 >= M) return;

    int half = lane >> 4;                        // 0 or 1
    int l16  = lane & 15;

    const float* arow = A + (size_t)(tile_m * 16 + l16) * K;
    const float* wcol = W + (size_t)(tile_n * 16 + l16);

    v8f c = {};
    #pragma unroll 4
    for (int k = 0; k < K; k += 4) {
        v2f a, b;
        int ka = k + 2 * half;
        a.x = arow[ka];
        a.y = arow[ka + 1];
        b.x = wcol[(size_t)ka * Nc];
        b.y = wcol[(size_t)(ka + 1) * Nc];
        // (neg_a, A, neg_b, B, c_mod, C, reuse_a, reuse_b)
        c = __builtin_amdgcn_wmma_f32_16x16x4_f32(false, a, false, b,
                                                  (short)0, c, false, false);
    }

    float* crow = C + (size_t)(tile_m * 16) * Nc + tile_n * 16;
    #pragma unroll
    for (int v = 0; v < 8; ++v) {
        crow[(size_t)(v + 8 * half) * Nc + l16] = c[v];
    }
}

// ---------------- edge gather/scale/scatter-add ----------------
// One wave per (edge or self-loop); lane moves 4 consecutive floats (coalesced 512B/row).

__global__ void gcn_edge_agg(const float* __restrict__ h, const int* __restrict__ src,
                             const int* __restrict__ dst, const float* __restrict__ dis,
                             float* __restrict__ out, int E, int N) {
    int idx  = (int)((blockIdx.x * blockDim.x + threadIdx.x) >> 5);
    int lane = threadIdx.x & 31;
    int total = E + N;
    if (idx >= total) return;

    int s, d;
    if (idx < E) { s = src[idx]; d = dst[idx]; }
    else         { s = idx - E; d = s; }        // self loop

    float nrm = dis[s] * dis[d];
    const float4* hs = (const float4*)(h + (size_t)s * FD);
    float4 v = hs[lane];
    float* o = out + (size_t)d * FD + lane * 4;
    unsafeAtomicAdd(o + 0, v.x * nrm);
    unsafeAtomicAdd(o + 1, v.y * nrm);
    unsafeAtomicAdd(o + 2, v.z * nrm);
    unsafeAtomicAdd(o + 3, v.w * nrm);
}

// ---------------- elementwise epilogues ----------------

__global__ void gcn_bias_relu(const float* __restrict__ in, const float* __restrict__ b,
                              float* __restrict__ out, int total) {
    int i = blockIdx.x * blockDim.x + threadIdx.x;
    if (i < total) {
        float v = in[i] + b[i & (FD - 1)];
        out[i] = v > 0.0f ? v : 0.0f;
    }
}

__global__ void gcn_bias_add(float* __restrict__ out, const float* __restrict__ b, int total) {
    int i = blockIdx.x * blockDim.x + threadIdx.x;
    if (i < total) out[i] += b[i & (FD - 1)];
}

// ---------------- driver ----------------

extern "C" void kernel_launch(void* const* d_in, const int* in_sizes, int n_in,
                              void* d_out, int out_size, void* d_ws, size_t ws_size,
                              hipStream_t stream) {
    const float* x  = (const float*)d_in[0];
    const int*   ei = (const int*)d_in[1];
    const float* W1 = (const float*)d_in[2];
    const float* b1 = (const float*)d_in[3];
    const float* W2 = (const float*)d_in[4];
    const float* b2 = (const float*)d_in[5];

    const int N = in_sizes[0] / FD;     // 50000
    const int E = in_sizes[1] / 2;      // 600000
    const int* src = ei;
    const int* dst = ei + E;

    float* ws   = (float*)d_ws;
    float* deg  = ws;                               // N
    float* dis  = ws + N;                           // N
    float* bufA = ws + 2 * (size_t)N;               // N*FD
    float* bufB = bufA + (size_t)N * FD;            // N*FD
    float* out  = (float*)d_out;

    const size_t feat_bytes = (size_t)N * FD * sizeof(float);
    const int total = N * FD;

    // normalization
    hipMemsetAsync(deg, 0, (size_t)N * sizeof(float), stream);
    gcn_deg_kernel<<<(E + 255) / 256, 256, 0, stream>>>(dst, deg, E);
    gcn_dis_kernel<<<(N + 255) / 256, 256, 0, stream>>>(deg, dis, N);

    // GEMM launch geometry: one wave per 16x16 tile
    const int tilesM  = (N + 15) / 16;              // 3125
    const int waves   = tilesM * (FD / 16);         // 25000
    const int gthreads = waves * 32;
    const int gblocks  = (gthreads + 255) / 256;

    // edge-agg geometry: one wave per edge/self-loop
    const int ethreads = (E + N) * 32;
    const int eblocks  = (ethreads + 255) / 256;

    // ---- layer 1 ----
    gcn_gemm_wmma<<<gblocks, 256, 0, stream>>>(x, W1, bufA, N);
    hipMemsetAsync(bufB, 0, feat_bytes, stream);
    gcn_edge_agg<<<eblocks, 256, 0, stream>>>(bufA, src, dst, dis, bufB, E, N);
    gcn_bias_relu<<<(total + 255) / 256, 256, 0, stream>>>(bufB, b1, bufA, total);

    // ---- layer 2 ----
    gcn_gemm_wmma<<<gblocks, 256, 0, stream>>>(bufA, W2, bufB, N);
    hipMemsetAsync(out, 0, feat_bytes, stream);
    gcn_edge_agg<<<eblocks, 256, 0, stream>>>(bufB, src, dst, dis, out, E, N);
    gcn_bias_add<<<(total + 255) / 256, 256, 0, stream>>>(out, b2, total);
}